// MBGCN_34067680592042
// MI455X (gfx1250) — compile-verified
//
#include <hip/hip_runtime.h>

// Problem constants (match reference)
#define NUU 100000
#define NII 50000
#define NN  150000
#define EE  1500000
#define DD  128
#define PITCH 132   // LDS row pitch (floats): conflict-free b64 fragment reads
#define RTILES 4    // 16-row tiles per block -> 64 rows/block

typedef __attribute__((ext_vector_type(2))) float v2f;
typedef __attribute__((ext_vector_type(8))) float v8f;

// ---------------------------------------------------------------------------
// WMMA fp32 GEMM: Out[M,128] = act( rowscale(A)[M,128] @ W[128,128] + bias
//                                   (+ A2[M,128] @ W2[128,128]) )
// 256 threads = 8 waves; wave w owns cols [16w,16w+16) for 4 row-tiles.
// W staged transposed in LDS (Wt[n][k]); A double-buffered in LDS.
// Inner loop: 2x ds_load b64 + v_wmma_f32_16x16x4_f32, K=128 in 32 steps.
// ---------------------------------------------------------------------------
__global__ __launch_bounds__(256) void mbgcn_gemm128(
    const float* __restrict__ A, const float* __restrict__ W,
    const float* __restrict__ bias,
    const float* __restrict__ A2, const float* __restrict__ W2,
    const float* __restrict__ rowscale,
    float* __restrict__ Out, int M, int relu)
{
  __shared__ float Wt[DD * PITCH];        // 67.6 KB, transposed weights
  __shared__ float At[2][16 * PITCH];     // 2 x 8.4 KB, A tiles (double buffer)

  const int t    = threadIdx.x;
  const int m0   = blockIdx.x * (16 * RTILES);
  const int lane = t & 31;
  const int nn   = lane & 15;        // N index in tile / A-row index
  const int ksel = (lane >> 4) * 2;  // K sub-select (0 | 2)
  const int n0   = (t >> 5) * 16;    // column tile of this wave
  const bool scaled = (rowscale != nullptr);

  v8f acc[RTILES];
  {
    const float bv = bias ? bias[n0 + nn] : 0.0f;
#pragma unroll
    for (int rt = 0; rt < RTILES; ++rt)
#pragma unroll
      for (int i = 0; i < 8; ++i) acc[rt][i] = bv;
  }

  // --- stage helpers ---
  auto stageW = [&](const float* __restrict__ Wsrc) {
    // coalesced read of W[k][n], write Wt[n*PITCH + k]
    for (int idx = t; idx < DD * DD; idx += 256) {
      const int k = idx >> 7, n = idx & (DD - 1);
      Wt[n * PITCH + k] = Wsrc[idx];
    }
  };
  auto stageA = [&](const float* __restrict__ Asrc, int rt, int buf, bool sc) {
    const int r  = t >> 4;
    const int c0 = (t & 15) * 8;
    int row = m0 + rt * 16 + r; if (row >= M) row = M - 1;
    const float* ap = Asrc + (size_t)row * DD + c0;
    const float s = sc ? rowscale[row] : 1.0f;
#pragma unroll
    for (int j = 0; j < 8; ++j) At[buf][r * PITCH + c0 + j] = ap[j] * s;
  };

  // --- pass 1: A @ W ---
  stageW(W);
  stageA(A, 0, 0, scaled);
  __syncthreads();
  int p = 0;
#pragma unroll
  for (int rt = 0; rt < RTILES; ++rt) {
    if (rt + 1 < RTILES) stageA(A, rt + 1, p ^ 1, scaled);  // overlap next tile
    const float* asp = &At[p][nn * PITCH + ksel];
    const float* wsp = &Wt[(n0 + nn) * PITCH + ksel];
#pragma unroll 8
    for (int k0 = 0; k0 < DD; k0 += 4) {
      v2f a; a.x = asp[k0]; a.y = asp[k0 + 1];
      v2f b; b.x = wsp[k0]; b.y = wsp[k0 + 1];
      acc[rt] = __builtin_amdgcn_wmma_f32_16x16x4_f32(false, a, false, b,
                                                      (short)0, acc[rt], false, false);
    }
    __syncthreads();
    p ^= 1;
  }

  // --- pass 2 (fused SAGE second operand): + A2 @ W2 ---
  if (A2) {
    stageW(W2);               // all Wt reads done at last barrier
    stageA(A2, 0, 0, false);
    __syncthreads();
    p = 0;
#pragma unroll
    for (int rt = 0; rt < RTILES; ++rt) {
      if (rt + 1 < RTILES) stageA(A2, rt + 1, p ^ 1, false);
      const float* asp = &At[p][nn * PITCH + ksel];
      const float* wsp = &Wt[(n0 + nn) * PITCH + ksel];
#pragma unroll 8
      for (int k0 = 0; k0 < DD; k0 += 4) {
        v2f a; a.x = asp[k0]; a.y = asp[k0 + 1];
        v2f b; b.x = wsp[k0]; b.y = wsp[k0 + 1];
        acc[rt] = __builtin_amdgcn_wmma_f32_16x16x4_f32(false, a, false, b,
                                                        (short)0, acc[rt], false, false);
      }
      __syncthreads();
      p ^= 1;
    }
  }

  // --- store: D layout VGPR i -> (M = i + 8*(lane/16), N = lane%16) ---
  const int mhi = (lane >> 4) * 8;
#pragma unroll
  for (int rt = 0; rt < RTILES; ++rt) {
    float* orow = Out + (size_t)(m0 + rt * 16) * DD + n0 + nn;
#pragma unroll
    for (int i = 0; i < 8; ++i) {
      const int m = i + mhi;
      if (m0 + rt * 16 + m < M) {
        float v = acc[rt][i];
        if (relu) v = fmaxf(v, 0.0f);
        orow[(size_t)m * DD] = v;
      }
    }
  }
}

// ---------------------------------------------------------------------------
// Per-behavior in-degree counting: cnt[b*N + dst] += 1
// ---------------------------------------------------------------------------
__global__ void mbgcn_edge_count(const int* __restrict__ ei,
                                 const int* __restrict__ et,
                                 float* __restrict__ cnt)
{
  const int e = blockIdx.x * blockDim.x + threadIdx.x;
  if (e >= EE) return;
  const int b = et[e];
  const int d = ei[EE + e];  // edge_index row 1 = dst
  unsafeAtomicAdd(&cnt[(size_t)b * NN + d], 1.0f);
}

__global__ void mbgcn_invden(float* __restrict__ cnt, int n)
{
  const int i = blockIdx.x * blockDim.x + threadIdx.x;
  if (i >= n) return;
  cnt[i] = 1.0f / fmaxf(cnt[i], 1.0f);
}

// ---------------------------------------------------------------------------
// Edge scatter: AGG[dst] += H[src] for edges of behavior bsel. 1 wave per edge,
// each lane moves 4 contiguous floats via native f32 global atomics.
// ---------------------------------------------------------------------------
__global__ __launch_bounds__(256) void mbgcn_edge_scatter(
    const int* __restrict__ ei, const int* __restrict__ et,
    const float* __restrict__ H, float* __restrict__ AGG, int bsel)
{
  const int lane = threadIdx.x & 31;
  const int e = blockIdx.x * 8 + (threadIdx.x >> 5);
  if (e >= EE) return;
  if (et[e] != bsel) return;
  const int s = ei[e];
  const int d = ei[EE + e];
  const float4 v = *(const float4*)(H + (size_t)s * DD + lane * 4);
  float* o = AGG + (size_t)d * DD + lane * 4;
  unsafeAtomicAdd(o + 0, v.x);
  unsafeAtomicAdd(o + 1, v.y);
  unsafeAtomicAdd(o + 2, v.z);
  unsafeAtomicAdd(o + 3, v.w);
}

// ---------------------------------------------------------------------------
// BatchNorm: stats (sum, sumsq per feature), finalize (scale/shift), apply.
// ---------------------------------------------------------------------------
__global__ __launch_bounds__(128) void mbgcn_bn_stats(
    const float* __restrict__ H, float* __restrict__ stats, int rows)
{
  const int f = threadIdx.x;  // feature 0..127 (coalesced across threads)
  float s = 0.0f, sq = 0.0f;
  for (int r = blockIdx.x; r < rows; r += gridDim.x) {
    const float v = H[(size_t)r * DD + f];
    s += v; sq += v * v;
  }
  unsafeAtomicAdd(&stats[f], s);
  unsafeAtomicAdd(&stats[DD + f], sq);
}

__global__ __launch_bounds__(128) void mbgcn_bn_finalize(
    float* __restrict__ stats, const float* __restrict__ gamma,
    const float* __restrict__ beta, float invN)
{
  const int f = threadIdx.x;
  const float mu  = stats[f] * invN;
  const float var = stats[DD + f] * invN - mu * mu;
  const float sc  = gamma[f] * rsqrtf(var + 1e-5f);
  stats[2 * DD + f] = sc;                 // scale
  stats[3 * DD + f] = beta[f] - mu * sc;  // shift
}

__global__ __launch_bounds__(256) void mbgcn_bn_apply(
    const float* __restrict__ H, const float* __restrict__ stats,
    const float* __restrict__ Xres, float* __restrict__ Out,
    int nelts, int relu)
{
  const int i = blockIdx.x * blockDim.x + threadIdx.x;
  if (i >= nelts) return;
  const int f = i & (DD - 1);
  float v = H[i] * stats[2 * DD + f] + stats[3 * DD + f];
  if (relu) v = fmaxf(v, 0.0f);
  if (Xres) v += Xres[i];
  Out[i] = v;
}

// ---------------------------------------------------------------------------
// Attention: logits[n,b] = dot(Q[n], K[n]); then softmax-weighted fusion.
// 1 wave per node.
// ---------------------------------------------------------------------------
__global__ __launch_bounds__(256) void mbgcn_attn_logits(
    const float* __restrict__ Q, const float* __restrict__ K,
    float* __restrict__ logits, int bsel)
{
  const int lane = threadIdx.x & 31;
  const int n = blockIdx.x * 8 + (threadIdx.x >> 5);
  if (n >= NN) return;
  const float4 q = *(const float4*)(Q + (size_t)n * DD + lane * 4);
  const float4 k = *(const float4*)(K + (size_t)n * DD + lane * 4);
  float p = q.x * k.x + q.y * k.y + q.z * k.z + q.w * k.w;
#pragma unroll
  for (int off = 16; off > 0; off >>= 1) p += __shfl_xor(p, off, 32);
  if (lane == 0) logits[(size_t)n * 3 + bsel] = p;
}

__global__ __launch_bounds__(256) void mbgcn_attn_fuse(
    const float* __restrict__ O0, const float* __restrict__ O1,
    const float* __restrict__ O2, const float* __restrict__ logits,
    float* __restrict__ Fused)
{
  const int lane = threadIdx.x & 31;
  const int n = blockIdx.x * 8 + (threadIdx.x >> 5);
  if (n >= NN) return;
  const float l0 = logits[(size_t)n * 3 + 0];
  const float l1 = logits[(size_t)n * 3 + 1];
  const float l2 = logits[(size_t)n * 3 + 2];
  const float mx = fmaxf(l0, fmaxf(l1, l2));
  float e0 = __expf(l0 - mx), e1 = __expf(l1 - mx), e2 = __expf(l2 - mx);
  const float inv = 1.0f / (e0 + e1 + e2);
  e0 *= inv; e1 *= inv; e2 *= inv;
  const size_t base = (size_t)n * DD + lane * 4;
  const float4 a = *(const float4*)(O0 + base);
  const float4 b = *(const float4*)(O1 + base);
  const float4 c = *(const float4*)(O2 + base);
  float4 r;
  r.x = e0 * a.x + e1 * b.x + e2 * c.x;
  r.y = e0 * a.y + e1 * b.y + e2 * c.y;
  r.z = e0 * a.z + e1 * b.z + e2 * c.z;
  r.w = e0 * a.w + e1 * b.w + e2 * c.w;
  *(float4*)(Fused + base) = r;
}

// ---------------------------------------------------------------------------
extern "C" void kernel_launch(void* const* d_in, const int* in_sizes, int n_in,
                              void* d_out, int out_size, void* d_ws, size_t ws_size,
                              hipStream_t stream) {
  const int*   edge_index  = (const int*)d_in[0];
  const int*   edge_type   = (const int*)d_in[1];
  const float* item_feats  = (const float*)d_in[2];
  const float* user_emb    = (const float*)d_in[3];
  const float* user_proj_W = (const float*)d_in[4];
  const float* user_proj_b = (const float*)d_in[5];
  const float* item_proj_W = (const float*)d_in[6];
  const float* item_proj_b = (const float*)d_in[7];
  const float* sage_Wl     = (const float*)d_in[8];   // [B,L,D,D]
  const float* sage_bl     = (const float*)d_in[9];   // [B,L,D]
  const float* sage_Wr     = (const float*)d_in[10];  // [B,L,D,D]
  const float* bn_gamma    = (const float*)d_in[11];  // [B,L,D]
  const float* bn_beta     = (const float*)d_in[12];
  const float* query_W     = (const float*)d_in[13];
  const float* query_b     = (const float*)d_in[14];
  const float* key_W       = (const float*)d_in[15];  // [B,D,D]
  const float* key_b       = (const float*)d_in[16];  // [B,D]
  const float* fuse_W      = (const float*)d_in[17];
  const float* fuse_b      = (const float*)d_in[18];
  const float* refine_W    = (const float*)d_in[19];
  const float* refine_b    = (const float*)d_in[20];

  const size_t ND = (size_t)NN * DD;
  float* ws    = (float*)d_ws;
  float* X     = ws;                 // [N,D]
  float* HA    = X    + ND;          // gemm out
  float* HB    = HA   + ND;          // activated layer buffer
  float* AGG   = HB   + ND;          // scatter accumulator
  float* O0    = AGG  + ND;          // per-behavior outputs
  float* O1    = O0   + ND;
  float* O2    = O1   + ND;
  float* QB    = O2   + ND;
  float* KB    = QB   + ND;
  float* LOGI  = KB   + ND;          // [N,3]
  float* INVD  = LOGI + (size_t)3 * NN;  // [3,N] (counts, then 1/deg)
  float* STATS = INVD + (size_t)3 * NN;  // 512 floats
  float* OUTB[3] = {O0, O1, O2};

  const dim3 blk256(256), blk128(128);
  const int ROWS = 16 * RTILES;
  const int gN  = (NN  + ROWS - 1) / ROWS;
  const int gNU = (NUU + ROWS - 1) / ROWS;
  const int gNI = (NII + ROWS - 1) / ROWS;

  // 1) per-behavior degrees -> inverse denominators
  hipMemsetAsync(INVD, 0, (size_t)3 * NN * sizeof(float), stream);
  mbgcn_edge_count<<<(EE + 255) / 256, blk256, 0, stream>>>(edge_index, edge_type, INVD);
  mbgcn_invden<<<(3 * NN + 255) / 256, blk256, 0, stream>>>(INVD, 3 * NN);

  // 2) initial node features X = [user_emb@Wu+bu ; item_feats@Wi+bi]
  mbgcn_gemm128<<<gNU, blk256, 0, stream>>>(
      user_emb, user_proj_W, user_proj_b, nullptr, nullptr, nullptr, X, NUU, 0);
  mbgcn_gemm128<<<gNI, blk256, 0, stream>>>(
      item_feats, item_proj_W, item_proj_b, nullptr, nullptr, nullptr,
      X + (size_t)NUU * DD, NII, 0);

  // 3) per-behavior 2-layer SAGE + BN (+ReLU) + residual
  for (int b = 0; b < 3; ++b) {
    const float* hcur = X;
    for (int l = 0; l < 2; ++l) {
      const size_t wi = ((size_t)b * 2 + l);
      hipMemsetAsync(AGG, 0, ND * sizeof(float), stream);
      mbgcn_edge_scatter<<<EE / 8, blk256, 0, stream>>>(edge_index, edge_type, hcur, AGG, b);
      // HA = (AGG/deg)@Wl + bl + hcur@Wr
      mbgcn_gemm128<<<gN, blk256, 0, stream>>>(
          AGG, sage_Wl + wi * DD * DD, sage_bl + wi * DD,
          hcur, sage_Wr + wi * DD * DD, INVD + (size_t)b * NN, HA, NN, 0);
      // BatchNorm (training-mode batch stats)
      hipMemsetAsync(STATS, 0, 2 * DD * sizeof(float), stream);
      mbgcn_bn_stats<<<1024, blk128, 0, stream>>>(HA, STATS, NN);
      mbgcn_bn_finalize<<<1, blk128, 0, stream>>>(STATS, bn_gamma + wi * DD,
                                                  bn_beta + wi * DD, 1.0f / NN);
      if (l == 0) {
        mbgcn_bn_apply<<<(int)((ND + 255) / 256), blk256, 0, stream>>>(
            HA, STATS, nullptr, HB, (int)ND, 1);
        hcur = HB;
      } else {
        mbgcn_bn_apply<<<(int)((ND + 255) / 256), blk256, 0, stream>>>(
            HA, STATS, X, OUTB[b], (int)ND, 0);  // residual x + h
      }
    }
  }

  // 4) attention fusion across behaviors
  mbgcn_gemm128<<<gN, blk256, 0, stream>>>(
      X, query_W, query_b, nullptr, nullptr, nullptr, QB, NN, 0);
  for (int b = 0; b < 3; ++b) {
    mbgcn_gemm128<<<gN, blk256, 0, stream>>>(
        OUTB[b], key_W + (size_t)b * DD * DD, key_b + (size_t)b * DD,
        nullptr, nullptr, nullptr, KB, NN, 0);
    mbgcn_attn_logits<<<(NN + 7) / 8, blk256, 0, stream>>>(QB, KB, LOGI, b);
  }
  mbgcn_attn_fuse<<<(NN + 7) / 8, blk256, 0, stream>>>(O0, O1, O2, LOGI, KB);

  // 5) fuse + refine (ReLU) -> d_out
  mbgcn_gemm128<<<gN, blk256, 0, stream>>>(
      KB, fuse_W, fuse_b, nullptr, nullptr, nullptr, QB, NN, 0);
  mbgcn_gemm128<<<gN, blk256, 0, stream>>>(
      QB, refine_W, refine_b, nullptr, nullptr, nullptr, (float*)d_out, NN, 1);
}